// CBFHalfspace_68917045231689
// MI455X (gfx1250) — compile-verified
//
#include <hip/hip_runtime.h>

typedef __attribute__((ext_vector_type(2))) float v2f;
typedef __attribute__((ext_vector_type(4))) float v4f;
typedef __attribute__((ext_vector_type(8))) float v8f;

#define TILES_PER_WAVE 8
#define WAVES_PER_BLOCK 8   // 256 threads = 8 wave32

// out[m] = [1+x0, 1-x0, 1+x1, 1-x1, 0,0,0,0]
// Computed per 16-state tile as D(16x16) = A(W',16x4) x B(x-tile,4x16) + C(bias)
// with V_WMMA_F32_16X16X4_F32 (full f32 -> bit-faithful to the f32 reference).
//
// Key layout tricks (zero EXEC churn in the hot loop):
//  * A columns K=2,3 are zero, so B's K2/K3 rows (lanes 16-31) are don't-care:
//    ALL lanes issue one unpredicated b64 load (upper lanes read x[n][2..3]).
//  * Output features 4..7 are exact +0.0 and upper lanes' d[0..3] are exact
//    +0.0 (zero A rows, zero C), so ALL lanes store one b128: lane L writes
//    out[row+L][0..3], lane L+16 writes out[row+L][4..7] -> 512 contiguous B.
__global__ __launch_bounds__(WAVES_PER_BLOCK * 32)
void cbf_halfspace_wmma_kernel(const float* __restrict__ x,
                               float* __restrict__ out,
                               int nrows)
{
    const int  lane   = threadIdx.x & 31;
    const int  laneLo = lane & 15;
    const bool lower  = lane < 16;
    const int  wave   = blockIdx.x * WAVES_PER_BLOCK + (threadIdx.x >> 5);

    // A-matrix (16x4 f32) = W': lanes 0-15 hold row M=lane {K0,K1},
    // lanes 16-31 hold {K2,K3} of the same rows (all zero).
    v2f a;
    a[0] = lower ? ((laneLo == 0) ? 1.0f : (laneLo == 1) ? -1.0f : 0.0f) : 0.0f;
    a[1] = lower ? ((laneLo == 2) ? 1.0f : (laneLo == 3) ? -1.0f : 0.0f) : 0.0f;

    // C bias: C[m][n] = (m < 4) ? 1 : 0.  VGPR j holds M=j (lo lanes) / j+8 (hi).
    v8f cb = {0.f, 0.f, 0.f, 0.f, 0.f, 0.f, 0.f, 0.f};
    if (lower) { cb[0] = 1.f; cb[1] = 1.f; cb[2] = 1.f; cb[3] = 1.f; }

    const int row0  = wave * (TILES_PER_WAVE * 16);
    const int myrow = row0 + laneLo;

    // One 64-bit per-lane address each, then pure immediate-offset stepping.
    const float* xp = x   + (size_t)myrow * 7 + (lower ? 0 : 2);  // 4B aligned
    float*       op = out + (size_t)myrow * 8 + (lower ? 0 : 4);  // 16B aligned

    if (row0 + TILES_PER_WAVE * 16 <= nrows) {
        // Fast path: load_b64 / wmma / store_b128 per tile, no predication.
#pragma unroll
        for (int t = 0; t < TILES_PER_WAVE; ++t) {
            v2f b;
            b[0] = xp[t * 112 + 0];   // 112 floats = 16 rows * 7 = 448B/tile
            b[1] = xp[t * 112 + 1];
            v8f d = __builtin_amdgcn_wmma_f32_16x16x4_f32(
                false, a, false, b, (short)0, cb, false, false);
            v4f s = {d[0], d[1], d[2], d[3]};
            __builtin_nontemporal_store(s, (v4f*)(op + t * 128)); // 512B/tile
        }
    } else {
        // Ragged tail (not taken for nrows = 4,194,304): per-lane guarded.
        for (int t = 0; t < TILES_PER_WAVE; ++t) {
            const int r0 = row0 + t * 16;
            if (r0 >= nrows) break;                 // wave-uniform
            const bool ok = (r0 + laneLo) < nrows;
            v2f b; b[0] = 0.0f; b[1] = 0.0f;
            if (ok) { b[0] = xp[t * 112 + 0]; b[1] = xp[t * 112 + 1]; }
            v8f d = __builtin_amdgcn_wmma_f32_16x16x4_f32(
                false, a, false, b, (short)0, cb, false, false);
            if (ok) {
                v4f s = {d[0], d[1], d[2], d[3]};
                *(v4f*)(op + t * 128) = s;
            }
        }
    }
}

extern "C" void kernel_launch(void* const* d_in, const int* in_sizes, int n_in,
                              void* d_out, int out_size, void* d_ws, size_t ws_size,
                              hipStream_t stream) {
    // setup_inputs order: x (B*7 f32), f (7 f32), g (14 f32). f and g contribute
    // exactly zero to the output (grad of the affine h is the zero vector).
    const float* x   = (const float*)d_in[0];
    float*       out = (float*)d_out;

    const int  nrows = in_sizes[0] / 7;                         // 4,194,304
    const long tiles = ((long)nrows + 15) / 16;                 // 262,144
    const long waves = (tiles + TILES_PER_WAVE - 1) / TILES_PER_WAVE;
    const int  blocks = (int)((waves + WAVES_PER_BLOCK - 1) / WAVES_PER_BLOCK);

    cbf_halfspace_wmma_kernel<<<blocks, WAVES_PER_BLOCK * 32, 0, stream>>>(x, out, nrows);
}